// VectorQuantizer_47218870452253
// MI455X (gfx1250) — compile-verified
//
#include <hip/hip_runtime.h>

typedef __attribute__((ext_vector_type(2))) float v2f;
typedef __attribute__((ext_vector_type(8))) float v8f;

#define K_CODES 1024
#define C_DIM   64
#define HW      1024      // 32*32 spatial
#define NHW     32768     // 32*32*32 rows
#define NCHW    2097152   // 32*64*32*32

// ---------------------------------------------------------------------------
// Kernel 1: e_sq[k] = sum_c embedding[k][c]^2
// ---------------------------------------------------------------------------
__global__ void vq_esq_kernel(const float* __restrict__ emb,
                              float* __restrict__ esq) {
    int k = blockIdx.x * blockDim.x + threadIdx.x;
    if (k >= K_CODES) return;
    const float4* row = (const float4*)(emb + (size_t)k * C_DIM);
    float s = 0.f;
#pragma unroll
    for (int i = 0; i < C_DIM / 4; ++i) {
        float4 q = row[i];
        s += q.x * q.x + q.y * q.y + q.z * q.z + q.w * q.w;
    }
    esq[k] = s;
}

// ---------------------------------------------------------------------------
// Kernel 2: one wave32 per tile of 16 z-rows.
//   - A (16x64 f32) held in 32 VGPRs for the whole tile
//   - loop 64 chunks of 16 codes; 16x v_wmma_f32_16x16x4_f32 per chunk
//   - running min/argmin of (e_sq[k] - 2*dot) per lane, xor-shuffle reduce
//   - gather winning code rows -> decoder_input, zq; index -> float out
// ---------------------------------------------------------------------------
__global__ __launch_bounds__(256)
void vq_assign_kernel(const float* __restrict__ feather,
                      const float* __restrict__ emb,
                      const float* __restrict__ esq,
                      float* __restrict__ dec_out,   // [N,C,H,W]
                      float* __restrict__ zq_out,    // [N,C,H,W]
                      float* __restrict__ idx_out)   // [N,1,H,W] as float
{
    __shared__ int kls[8 * 16];   // per-wave winning code per row

    const int lane = threadIdx.x & 31;
    const int wave = threadIdx.x >> 5;
    const int tile = blockIdx.x * 8 + wave;      // 2048 tiles total
    const int row0 = tile * 16;                  // first z-row of tile
    const int n    = row0 / HW;                  // batch index (tile never crosses n)
    const int hw0  = row0 % HW;                  // spatial base
    const int m    = lane & 15;                  // row within tile / code within chunk
    const int hi   = lane >> 4;                  // half-wave: K-pair select

    // ---- load A once: lane holds, for K-step kk, channels (4kk+2hi, 4kk+2hi+1)
    // of row m.  z[row][c] = feather[n*C*HW + c*HW + hw0 + m]
    const float* fbase = feather + (size_t)n * (C_DIM * HW) + hw0 + m;
    v2f a[16];
#pragma unroll
    for (int kk = 0; kk < 16; ++kk) {
        int c = 4 * kk + 2 * hi;
        a[kk].x = fbase[(size_t)c * HW];
        a[kk].y = fbase[(size_t)(c + 1) * HW];
    }

    float best[8];
    int   bidx[8];
#pragma unroll
    for (int i = 0; i < 8; ++i) { best[i] = 3.4e38f; bidx[i] = 0; }

    // ---- main loop: 64 chunks of 16 codes
    for (int k0 = 0; k0 < K_CODES; k0 += 16) {
        // B 4x16 f32: lane supplies channels (4kk+2hi, +1) of code (k0+m)
        const float* ebase = emb + (size_t)(k0 + m) * C_DIM + 2 * hi;
        v8f acc = {};
#pragma unroll
        for (int kk = 0; kk < 16; ++kk) {
            v2f b = *(const v2f*)(ebase + 4 * kk);
            acc = __builtin_amdgcn_wmma_f32_16x16x4_f32(
                false, a[kk], false, b, (short)0, acc, false, false);
        }
        // dist = e_sq[k] - 2*dot  (z_sq is constant per row -> same argmin)
        float es = esq[k0 + m];
#pragma unroll
        for (int i = 0; i < 8; ++i) {
            float d = es - 2.0f * acc[i];
            if (d < best[i]) { best[i] = d; bidx[i] = k0 + m; }
        }
    }

    // ---- reduce argmin across the 16 lanes of each half (codes mod 16)
#pragma unroll
    for (int s = 1; s < 16; s <<= 1) {
#pragma unroll
        for (int i = 0; i < 8; ++i) {
            float ob = __shfl_xor(best[i], s, 32);
            int   oi = __shfl_xor(bidx[i], s, 32);
            if (ob < best[i] || (ob == best[i] && oi < bidx[i])) {
                best[i] = ob; bidx[i] = oi;
            }
        }
    }

    // slots 0..7 of lanes 0-15 are rows 0..7; of lanes 16-31 are rows 8..15
    if (lane == 0 || lane == 16) {
#pragma unroll
        for (int i = 0; i < 8; ++i) kls[wave * 16 + hi * 8 + i] = bidx[i];
    }
    __syncthreads();
    const int kst = kls[wave * 16 + m];   // winning code for row m

    // ---- nearest index (int -> float), one per row
    if (lane < 16) idx_out[row0 + m] = (float)kst;

    // ---- gather code vector; 16 consecutive floats per (channel, tile) store
    const size_t obase = (size_t)n * (C_DIM * HW) + hw0;
    const float* erow  = emb + (size_t)kst * C_DIM;
#pragma unroll
    for (int c = 0; c < C_DIM; c += 2) {
        int cc  = c + hi;                 // lanes 0-15 -> c, lanes 16-31 -> c+1
        float v = erow[cc];
        size_t o = obase + (size_t)cc * HW + m;
        dec_out[o] = v;                   // decoder_input == zq numerically
        zq_out[o]  = v;
    }
}

// ---------------------------------------------------------------------------
extern "C" void kernel_launch(void* const* d_in, const int* in_sizes, int n_in,
                              void* d_out, int out_size, void* d_ws, size_t ws_size,
                              hipStream_t stream) {
    const float* feather = (const float*)d_in[0];   // [32,64,32,32]
    const float* emb     = (const float*)d_in[1];   // [1024,64]

    float* dec_out = (float*)d_out;                 // [32,64,32,32]
    float* zq_out  = dec_out + NCHW;                // [32,64,32,32]
    float* idx_out = zq_out + NCHW;                 // [32,1,32,32]

    float* esq = (float*)d_ws;                      // 1024 floats scratch

    vq_esq_kernel<<<K_CODES / 256, 256, 0, stream>>>(emb, esq);
    vq_assign_kernel<<<NHW / (16 * 8), 256, 0, stream>>>(
        feather, emb, esq, dec_out, zq_out, idx_out);
}